// GAT_28398323761834
// MI455X (gfx1250) — compile-verified
//
#include <hip/hip_runtime.h>
#include <hip/hip_bf16.h>
#include <math.h>

#define NEG_SLOPE 0.2f

typedef float v2f __attribute__((ext_vector_type(2)));
typedef float v8f __attribute__((ext_vector_type(8)));

// ---- order-preserving float<->uint encoding for atomicMax-based segment max ----
__device__ __forceinline__ unsigned int fenc(float f) {
    unsigned int u = __float_as_uint(f);
    return (u & 0x80000000u) ? ~u : (u | 0x80000000u);
}
__device__ __forceinline__ float fdec(unsigned int u) {
    return __uint_as_float((u & 0x80000000u) ? (u & 0x7FFFFFFFu) : ~u);
}
#define ENC_NEG_INF 0x007FFFFFu   // fenc(-inf)

__device__ __forceinline__ float lrelu(float v) {
    return v > 0.0f ? v : v * NEG_SLOPE;
}

// =====================================================================
// Layer 1 GEMM: out = x(N x 32) @ W(32 x 128) + b, via V_WMMA_F32_16X16X4_F32
// block = 256 threads = 8 waves; wave w -> col tile w*16; blockIdx.x -> row tile
// blockIdx.y: 0 = (Wl,bl)->xl   1 = (Wr,br)->xr
// =====================================================================
__global__ __launch_bounds__(256) void gat_gemm1_wmma(
    const float* __restrict__ x,
    const float* __restrict__ Wl, const float* __restrict__ bl,
    const float* __restrict__ Wr, const float* __restrict__ br,
    float* __restrict__ xl, float* __restrict__ xr, int n)
{
    const int lane = threadIdx.x & 31;
    const int wave = threadIdx.x >> 5;
    const int m0   = blockIdx.x * 16;
    const bool right = (blockIdx.y != 0);
    const float* __restrict__ W    = right ? Wr : Wl;
    const float* __restrict__ bias = right ? br : bl;
    float* __restrict__ out        = right ? xr : xl;

    const int half = lane >> 4;          // 0: lanes 0-15, 1: lanes 16-31
    const int l15  = lane & 15;
    const int kb   = half * 2;           // K sub-offset 0 or 2
    int mrow = m0 + l15; if (mrow >= n) mrow = n - 1;   // clamp (EXEC stays all-1)
    const int ncol = wave * 16 + l15;

    v8f acc = {};
#pragma unroll
    for (int kk = 0; kk < 32; kk += 4) {
        v2f a, b;
        a.x = x[(size_t)mrow * 32 + kk + kb];
        a.y = x[(size_t)mrow * 32 + kk + kb + 1];
        b.x = W[(size_t)(kk + kb)     * 128 + ncol];
        b.y = W[(size_t)(kk + kb + 1) * 128 + ncol];
        acc = __builtin_amdgcn_wmma_f32_16x16x4_f32(
                  false, a, false, b, (short)0, acc, false, false);
    }
    const float bv = bias[ncol];
    float* op = out + (size_t)(m0 + half * 8) * 128 + ncol;
    if (m0 + 16 <= n) {
        // whole tile in range (always true when n % 16 == 0): straight-line stores
#pragma unroll
        for (int r = 0; r < 8; ++r) op[(size_t)r * 128] = acc[r] + bv;
    } else {
#pragma unroll
        for (int r = 0; r < 8; ++r) {
            const int m = m0 + r + half * 8;
            if (m < n) out[(size_t)m * 128 + ncol] = acc[r] + bv;
        }
    }
}

// ===================== init encoded-max buffers ======================
__global__ void gat_init_mx(unsigned int* mx1, int n1, unsigned int* mx2, int n2)
{
    int i = blockIdx.x * blockDim.x + threadIdx.x;
    if (i < n1) mx1[i] = ENC_NEG_INF;
    if (i < n2) mx2[i] = ENC_NEG_INF;
}

// ============ layer-1 per-edge logits + segment max (thread/edge) ============
__global__ __launch_bounds__(256) void gat_edge_logits1(
    const float* __restrict__ xl, const float* __restrict__ xr,
    const int* __restrict__ src, const int* __restrict__ dst,
    const float* __restrict__ att,        // 8x16
    float* __restrict__ logits,           // Etot x 8
    unsigned int* __restrict__ mx,        // N x 8 (encoded)
    int E, int Etot)
{
    int e = blockIdx.x * blockDim.x + threadIdx.x;
    if (e >= Etot) return;
    const int s = (e < E) ? src[e] : (e - E);
    const int d = (e < E) ? dst[e] : (e - E);
    const float4* ps = (const float4*)(xl + (size_t)s * 128);
    const float4* pd = (const float4*)(xr + (size_t)d * 128);
    // overlap the gather latency: prefetch both rows (global_prefetch_b8)
    __builtin_prefetch(ps, 0, 0);
    __builtin_prefetch(pd, 0, 0);
    __builtin_prefetch((const char*)ps + 256, 0, 0);
    __builtin_prefetch((const char*)pd + 256, 0, 0);
    const float4* pa = (const float4*)att;
#pragma unroll
    for (int h = 0; h < 8; ++h) {
        float acc = 0.0f;
#pragma unroll
        for (int q = 0; q < 4; ++q) {
            float4 a = ps[h * 4 + q];
            float4 b = pd[h * 4 + q];
            float4 w = pa[h * 4 + q];
            acc += lrelu(a.x + b.x) * w.x + lrelu(a.y + b.y) * w.y +
                   lrelu(a.z + b.z) * w.z + lrelu(a.w + b.w) * w.w;
        }
        logits[(size_t)e * 8 + h] = acc;
        atomicMax(&mx[(size_t)d * 8 + h], fenc(acc));
    }
}

// ============ layer-1 exp + segment sum (thread per edge*head) ============
__global__ __launch_bounds__(256) void gat_edge_exp1(
    float* __restrict__ logits,           // in: logit, out: exp
    const unsigned int* __restrict__ mx,
    float* __restrict__ den,              // N x 8 (zeroed)
    const int* __restrict__ dst, int E, int Etot)
{
    int i = blockIdx.x * blockDim.x + threadIdx.x;
    if (i >= Etot * 8) return;
    const int e = i >> 3, h = i & 7;
    const int d = (e < E) ? dst[e] : (e - E);
    const float ex = __expf(logits[i] - fdec(mx[(size_t)d * 8 + h]));
    logits[i] = ex;
    atomicAdd(&den[(size_t)d * 8 + h], ex);
}

// ============ layer-1 weighted scatter-add (wave per edge) ============
__global__ __launch_bounds__(256) void gat_edge_aggr1(
    const float* __restrict__ xl, const float* __restrict__ ex,
    const float* __restrict__ den,
    const int* __restrict__ src, const int* __restrict__ dst,
    float* __restrict__ hacc,             // N x 128 (zeroed)
    int E, int Etot)
{
    const int e = (int)((blockIdx.x * (size_t)blockDim.x + threadIdx.x) >> 5);
    const int lane = threadIdx.x & 31;
    if (e >= Etot) return;
    const int s = (e < E) ? src[e] : (e - E);
    const int d = (e < E) ? dst[e] : (e - E);
    const float4* psrc = (const float4*)(xl + (size_t)s * 128) + lane;
    __builtin_prefetch(psrc, 0, 0);       // hide gather latency behind alpha math
    const int h = lane >> 2;              // 4 lanes per head (16 ch / 4 ch-per-lane)
    const float alpha = ex[(size_t)e * 8 + h] / den[(size_t)d * 8 + h];
    const float4 v = *psrc;
    float* o = hacc + (size_t)d * 128 + lane * 4;
    atomicAdd(o + 0, alpha * v.x);
    atomicAdd(o + 1, alpha * v.y);
    atomicAdd(o + 2, alpha * v.z);
    atomicAdd(o + 3, alpha * v.w);
}

// ============ elu(h + bias1), in place ============
__global__ __launch_bounds__(256) void gat_elu_bias(
    float* __restrict__ h, const float* __restrict__ bias1, int total)
{
    int i = blockIdx.x * blockDim.x + threadIdx.x;
    if (i >= total) return;
    float v = h[i] + bias1[i & 127];
    h[i] = v > 0.0f ? v : (__expf(v) - 1.0f);
}

// ============ layer-2 matvec: xl2/xr2 = h(N x 128) @ W(128x1) + b ============
__global__ __launch_bounds__(256) void gat_matvec2(
    const float* __restrict__ h,
    const float* __restrict__ Wl2, const float* __restrict__ bl2,
    const float* __restrict__ Wr2, const float* __restrict__ br2,
    float* __restrict__ xl2, float* __restrict__ xr2, int n)
{
    int i = blockIdx.x * blockDim.x + threadIdx.x;
    if (i >= n) return;
    const float4* ph = (const float4*)(h + (size_t)i * 128);
    const float4* pl = (const float4*)Wl2;
    const float4* pr = (const float4*)Wr2;
    float al = bl2[0], ar = br2[0];
#pragma unroll
    for (int q = 0; q < 32; ++q) {
        float4 v = ph[q], wl = pl[q], wr = pr[q];
        al += v.x * wl.x + v.y * wl.y + v.z * wl.z + v.w * wl.w;
        ar += v.x * wr.x + v.y * wr.y + v.z * wr.z + v.w * wr.w;
    }
    xl2[i] = al; xr2[i] = ar;
}

// ============ layer-2 logits + segment max ============
__global__ __launch_bounds__(256) void gat_edge_logits2(
    const float* __restrict__ xl2, const float* __restrict__ xr2,
    const int* __restrict__ src, const int* __restrict__ dst,
    const float* __restrict__ att2,
    float* __restrict__ logit2, unsigned int* __restrict__ mx2,
    int E, int Etot)
{
    int e = blockIdx.x * blockDim.x + threadIdx.x;
    if (e >= Etot) return;
    const int s = (e < E) ? src[e] : (e - E);
    const int d = (e < E) ? dst[e] : (e - E);
    const float lg = lrelu(xl2[s] + xr2[d]) * att2[0];
    logit2[e] = lg;
    atomicMax(&mx2[d], fenc(lg));
}

__global__ __launch_bounds__(256) void gat_edge_exp2(
    float* __restrict__ logit2, const unsigned int* __restrict__ mx2,
    float* __restrict__ den2, const int* __restrict__ dst, int E, int Etot)
{
    int e = blockIdx.x * blockDim.x + threadIdx.x;
    if (e >= Etot) return;
    const int d = (e < E) ? dst[e] : (e - E);
    const float ex = __expf(logit2[e] - fdec(mx2[d]));
    logit2[e] = ex;
    atomicAdd(&den2[d], ex);
}

__global__ __launch_bounds__(256) void gat_edge_aggr2(
    const float* __restrict__ xl2, const float* __restrict__ ex2,
    const float* __restrict__ den2,
    const int* __restrict__ src, const int* __restrict__ dst,
    float* __restrict__ out, int E, int Etot)
{
    int e = blockIdx.x * blockDim.x + threadIdx.x;
    if (e >= Etot) return;
    const int s = (e < E) ? src[e] : (e - E);
    const int d = (e < E) ? dst[e] : (e - E);
    atomicAdd(&out[d], (ex2[e] / den2[d]) * xl2[s]);
}

__global__ __launch_bounds__(256) void gat_bias_out(
    float* __restrict__ out, const float* __restrict__ bias2, int n)
{
    int i = blockIdx.x * blockDim.x + threadIdx.x;
    if (i < n) out[i] += bias2[0];
}

// =====================================================================
extern "C" void kernel_launch(void* const* d_in, const int* in_sizes, int n_in,
                              void* d_out, int out_size, void* d_ws, size_t ws_size,
                              hipStream_t stream)
{
    const float* x     = (const float*)d_in[0];
    const int*   src   = (const int*)  d_in[1];
    const int*   dst   = (const int*)  d_in[2];
    const float* Wl1   = (const float*)d_in[3];
    const float* bl1   = (const float*)d_in[4];
    const float* Wr1   = (const float*)d_in[5];
    const float* br1   = (const float*)d_in[6];
    const float* att1  = (const float*)d_in[7];
    const float* bias1 = (const float*)d_in[8];
    const float* Wl2   = (const float*)d_in[9];
    const float* bl2   = (const float*)d_in[10];
    const float* Wr2   = (const float*)d_in[11];
    const float* br2   = (const float*)d_in[12];
    const float* att2  = (const float*)d_in[13];
    const float* bias2 = (const float*)d_in[14];

    const int n    = in_sizes[0] / 32;   // 50000
    const int E    = in_sizes[1];        // 800000
    const int Etot = E + n;              // with self-loops

    // ---- workspace carve-up (floats) ----
    float* base = (float*)d_ws;
    size_t off = 0;
    float* xl1      = base + off; off += (size_t)n * 128;
    float* xr1      = base + off; off += (size_t)n * 128;  // reused as hacc/h later
    float* logits1  = base + off; off += (size_t)Etot * 8; // reused as ex1
    unsigned int* mx1 = (unsigned int*)(base + off); off += (size_t)n * 8;
    float* den1     = base + off; off += (size_t)n * 8;
    float* xl2      = base + off; off += n;
    float* xr2      = base + off; off += n;
    float* logit2   = base + off; off += Etot;             // reused as ex2
    unsigned int* mx2 = (unsigned int*)(base + off); off += n;
    float* den2     = base + off; off += n;
    float* hacc     = xr1;                                 // alias (xr1 dead after exp1)
    float* outp     = (float*)d_out;

    // ---- zero/init accumulators (stream-ordered; graph-capture safe) ----
    hipMemsetAsync(den1, 0, (size_t)n * 8 * sizeof(float), stream);
    hipMemsetAsync(den2, 0, (size_t)n * sizeof(float), stream);
    hipMemsetAsync(outp, 0, (size_t)n * sizeof(float), stream);
    {
        int tot = n * 8;
        gat_init_mx<<<(tot + 255) / 256, 256, 0, stream>>>(mx1, tot, mx2, n);
    }

    // ---- layer 1 ----
    gat_gemm1_wmma<<<dim3((n + 15) / 16, 2), 256, 0, stream>>>(
        x, Wl1, bl1, Wr1, br1, xl1, xr1, n);
    gat_edge_logits1<<<(Etot + 255) / 256, 256, 0, stream>>>(
        xl1, xr1, src, dst, att1, logits1, mx1, E, Etot);
    gat_edge_exp1<<<(Etot * 8 + 255) / 256, 256, 0, stream>>>(
        logits1, mx1, den1, dst, E, Etot);
    hipMemsetAsync(hacc, 0, (size_t)n * 128 * sizeof(float), stream); // xr1 now dead
    {
        size_t waves = (size_t)Etot;                   // one wave per edge
        size_t blocks = (waves * 32 + 255) / 256;
        gat_edge_aggr1<<<(int)blocks, 256, 0, stream>>>(
            xl1, logits1, den1, src, dst, hacc, E, Etot);
    }
    gat_elu_bias<<<((n * 128) + 255) / 256, 256, 0, stream>>>(hacc, bias1, n * 128);

    // ---- layer 2 ----
    gat_matvec2<<<(n + 255) / 256, 256, 0, stream>>>(
        hacc, Wl2, bl2, Wr2, br2, xl2, xr2, n);
    gat_edge_logits2<<<(Etot + 255) / 256, 256, 0, stream>>>(
        xl2, xr2, src, dst, att2, logit2, mx2, E, Etot);
    gat_edge_exp2<<<(Etot + 255) / 256, 256, 0, stream>>>(
        logit2, mx2, den2, dst, E, Etot);
    gat_edge_aggr2<<<(Etot + 255) / 256, 256, 0, stream>>>(
        xl2, logit2, den2, src, dst, outp, E, Etot);
    gat_bias_out<<<(n + 255) / 256, 256, 0, stream>>>(outp, bias2, n);
}